// PNALayer_73297911873708
// MI455X (gfx1250) — compile-verified
//
#include <hip/hip_runtime.h>
#include <hip/hip_bf16.h>
#include <math.h>

// ---------------------------------------------------------------------------
// PNA layer for MI455X (gfx1250, wave32, WMMA, async LDS copies).
// Phase 1: edge pretrans GEMM (gathered) + relu + atomic scatter aggregation
// Phase 2: per-node finalize (mean/std/max/min + degree scalers)
// Phase 3: node posttrans 2-layer MLP GEMMs + residual
// fp32 WMMA (V_WMMA_F32_16X16X4_F32): workload is memory/atomic bound, so
// fp32 matrix ops cost nothing vs bf16 and preserve reference numerics.
// Weights are pre-interleaved into K-pair float2 so each WMMA B operand is a
// single 8B LDS load. Tiles staged with GLOBAL_LOAD_ASYNC_TO_LDS_B128 via
// inline asm (bypasses the clang builtin signature), ordered by
// s_wait_asynccnt per the CDNA5 ASYNCcnt discipline.
// ---------------------------------------------------------------------------

#define N_NODES   50000
#define N_EDGES   600000
#define IN_DIM    128
#define EDGE_DIM  64
#define OUT_DIM   128
#define AVG_D_LOG 2.5f
#define PNA_EPS   1e-5f

typedef __attribute__((ext_vector_type(2))) float v2f;
typedef __attribute__((ext_vector_type(8))) float v8f;

// 16-byte global -> LDS async copy (ASYNCcnt-tracked, no VGPR round trip).
// Flat shared pointers carry the LDS byte offset in their low 32 bits
// (ISA 10.2 aperture mapping), so truncation yields the VDST LDS address.
__device__ __forceinline__ void cp16(void* lds, const void* gmem) {
  unsigned loff = (unsigned)(unsigned long long)lds;
  asm volatile("global_load_async_to_lds_b128 %0, %1, off"
               :
               : "v"(loff), "v"(gmem)
               : "memory");
}
// Wait for all outstanding async LDS copies from this wave.
__device__ __forceinline__ void cp_commit() {
  asm volatile("s_wait_asynccnt 0" ::: "memory");
}

__device__ __forceinline__ v8f wmma_f32x4(v2f a, v2f b, v8f c) {
  // D = A(16x4) * B(4x16) + C(16x16), fp32, wave32.
  return __builtin_amdgcn_wmma_f32_16x16x4_f32(
      /*neg_a=*/false, a, /*neg_b=*/false, b,
      /*c_mod=*/(short)0, c, /*reuse_a=*/false, /*reuse_b=*/false);
}

// ---------------------------------------------------------------------------
// Kernel W: interleave weight rows into K-pair float2: Wp[kk][n] = {W[2kk][n],
// W[2kk+1][n]}. Makes each WMMA B fragment one contiguous 8B load.
// ---------------------------------------------------------------------------
__global__ __launch_bounds__(256) void pna_interleave_w(
    const float* __restrict__ W, float2* __restrict__ Wp, int Krows) {
  int i = blockIdx.x * 256 + threadIdx.x;         // over (Krows/2)*128
  int total = (Krows >> 1) * 128;
  if (i >= total) return;
  int kk = i >> 7, n = i & 127;
  float2 v;
  v.x = W[(size_t)(2 * kk) * 128 + n];
  v.y = W[(size_t)(2 * kk + 1) * 128 + n];
  Wp[i] = v;
}

// ---------------------------------------------------------------------------
// Kernel 0: init accumulators. acc layout: [N][4][128] = sum|maxbits|minbits|sumsq
// max init = 0 (messages are post-relu, >= 0), min init = +inf bits.
// ---------------------------------------------------------------------------
__global__ __launch_bounds__(256) void pna_init(unsigned* __restrict__ acc,
                                                float* __restrict__ deg) {
  long i = (long)blockIdx.x * 256 + threadIdx.x;
  long total = (long)N_NODES * 512;
  if (i < total) {
    int slot = (int)((i >> 7) & 3);
    acc[i] = (slot == 2) ? 0x7F800000u : 0u;      // min slot -> +inf bits
  }
  if (i < N_NODES) deg[i] = 0.0f;
}

// ---------------------------------------------------------------------------
// Kernel 1: edge pretrans + scatter aggregation.
// Tile: 64 edges x 128 outputs, 256 threads = 8 waves.
// K = 320 in 5 chunks of 64 (src feats | dst feats | edge feats).
// sB2 row stride = 144 float2 = 288 dwords (== 32 mod 64): lane halves hit
// disjoint LDS bank halves -> conflict-free 8B B-fragment loads.
// ---------------------------------------------------------------------------
__global__ __launch_bounds__(256) void pna_edge(
    const float* __restrict__ x, const float* __restrict__ ef,
    const int* __restrict__ src, const int* __restrict__ dst,
    const float2* __restrict__ Wp2, const float* __restrict__ bp,
    float* __restrict__ acc, float* __restrict__ deg) {
  __shared__ __align__(16) float  sA[64][68];     // edges x k-chunk
  __shared__ __align__(16) float2 sB2[32][144];   // k-pairs x 128 (+pad)
  __shared__ int sSrc[64];
  __shared__ int sDst[64];

  const int tid = threadIdx.x;
  const int e0 = blockIdx.x * 64;

  if (tid < 64) {
    int s = src[e0 + tid];
    int d = dst[e0 + tid];
    sSrc[tid] = s;
    sDst[tid] = d;
    atomicAdd(&deg[d], 1.0f);                     // degree count
  }
  __syncthreads();

  const int lane  = tid & 31;
  const int wid   = tid >> 5;
  const int mbase = (wid & 3) << 4;               // 0,16,32,48
  const int nbase = (wid >> 2) << 6;              // 0 or 64
  const int ml    = lane & 15;
  const int kp    = lane >> 4;                    // K half select

  v8f accf[4];
  #pragma unroll
  for (int f = 0; f < 4; ++f) accf[f] = (v8f){0, 0, 0, 0, 0, 0, 0, 0};

  #pragma unroll 1
  for (int c = 0; c < 5; ++c) {
    // ---- stage A chunk: 64 edges x 64 k, verbatim gather (async 16B) ----
    for (int i = tid; i < 64 * 16; i += 256) {
      int e  = i >> 4;
      int k4 = (i & 15) << 2;
      const float* p;
      if (c < 2)       p = &x[(size_t)sSrc[e] * IN_DIM + c * 64 + k4];
      else if (c < 4)  p = &x[(size_t)sDst[e] * IN_DIM + (c - 2) * 64 + k4];
      else             p = &ef[(size_t)(e0 + e) * EDGE_DIM + k4];
      cp16(&sA[e][k4], p);
    }
    // ---- stage B chunk: interleaved W_pre pairs, verbatim (async 16B) ----
    for (int i = tid; i < 32 * 64; i += 256) {    // 64 x 16B per row
      int kk = i >> 6;
      int q  = (i & 63) << 1;                     // float2 index within row
      cp16(&sB2[kk][q], &Wp2[(size_t)(c * 32 + kk) * 128 + q]);
    }
    if (c + 1 < 5)
      __builtin_prefetch(&Wp2[(size_t)(c + 1) * 32 * 128 + tid * 16], 0, 1);
    cp_commit();
    __syncthreads();

    // ---- WMMA over this K chunk ----
    #pragma unroll
    for (int k0 = 0; k0 < 64; k0 += 4) {
      v2f a = *(const v2f*)&sA[mbase + ml][k0 + 2 * kp];
      int kkB = (k0 >> 1) + kp;
      #pragma unroll
      for (int f = 0; f < 4; ++f) {
        v2f b = *(const v2f*)&sB2[kkB][nbase + f * 16 + ml];
        accf[f] = wmma_f32x4(a, b, accf[f]);
      }
    }
    __syncthreads();
  }

  // ---- epilogue: relu(bias) + 4 atomic aggregation streams ----
  // C frag layout: reg r, lane L -> m = r + 8*(L>>4), n = L&15
  #pragma unroll
  for (int f = 0; f < 4; ++f) {
    int n = nbase + f * 16 + ml;
    float bb = bp[n];
    #pragma unroll
    for (int r = 0; r < 8; ++r) {
      float v = fmaxf(accf[f][r] + bb, 0.0f);
      int m = mbase + r + 8 * kp;
      int node = sDst[m];
      float* row = acc + (size_t)node * 512 + n;
      atomicAdd(row, v);                                          // sum
      atomicMax((unsigned*)(row + 128), __float_as_uint(v));      // max (v>=0)
      atomicMin((unsigned*)(row + 256), __float_as_uint(v));      // min (v>=0)
      atomicAdd(row + 384, v * v);                                // sumsq
    }
  }
}

// ---------------------------------------------------------------------------
// Kernel 2: finalize agg in place: [sum|max|min|sumsq] -> [mean|max|min|std],
// masked by has_edges; also per-node amplification/attenuation scalers.
// ---------------------------------------------------------------------------
__global__ __launch_bounds__(256) void pna_finalize(
    float* __restrict__ acc, const float* __restrict__ deg,
    float* __restrict__ amp, float* __restrict__ att) {
  int g = blockIdx.x * 256 + threadIdx.x;
  int n = g >> 7;
  int d = g & 127;
  if (n >= N_NODES) return;
  float dg = deg[n];
  float* row = acc + (size_t)n * 512 + d;
  if (dg > 0.0f) {
    float inv  = 1.0f / dg;                       // safe_deg == deg here
    float mean = row[0] * inv;
    float var  = fmaxf(row[384] * inv - mean * mean, 0.0f);
    row[0]   = mean;
    row[384] = sqrtf(var + PNA_EPS);
    // row[128] (max) and row[256] (min) already hold correct float bits
  } else {
    row[0] = 0.0f; row[128] = 0.0f; row[256] = 0.0f; row[384] = 0.0f;
  }
  if (d == 0) {
    float logd = logf(dg + 1.0f);
    amp[n] = logd * (1.0f / AVG_D_LOG);
    att[n] = (dg > 0.0f) ? (AVG_D_LOG / fmaxf(logd, PNA_EPS)) : 0.0f;
  }
}

// ---------------------------------------------------------------------------
// Kernel 3: node posttrans. Tile 64 nodes x 128 out, 256 threads = 8 waves.
// h (K=1664) built on the fly in 26 chunks of 64:
//   c<2: node_feat; c<10: agg; c<18: agg*amp; c<26: agg*att
// GEMM1 vs W_post1 -> relu -> LDS (reusing sB2 storage); GEMM2 vs
// pre-interleaved W_post2 (8B global loads, L2 resident); + bias + residual.
// ---------------------------------------------------------------------------
__global__ __launch_bounds__(256) void pna_node(
    const float* __restrict__ x, const float* __restrict__ agg,
    const float* __restrict__ amp, const float* __restrict__ att,
    const float2* __restrict__ W1p, const float* __restrict__ b1,
    const float2* __restrict__ W2p, const float* __restrict__ b2,
    float* __restrict__ out) {
  __shared__ __align__(16) float  sA[64][68];     // nodes x k-chunk
  __shared__ __align__(16) float2 sB2[32][144];   // W1 k-pairs x 128 (+pad)
  __shared__ float sAmp[64];
  __shared__ float sAtt[64];

  const int tid = threadIdx.x;
  const int n0  = blockIdx.x * 64;

  if (tid < 64) {
    int node = min(n0 + tid, N_NODES - 1);
    sAmp[tid] = amp[node];
    sAtt[tid] = att[node];
  }
  __syncthreads();

  const int lane  = tid & 31;
  const int wid   = tid >> 5;
  const int mbase = (wid & 3) << 4;
  const int nbase = (wid >> 2) << 6;
  const int ml    = lane & 15;
  const int kp    = lane >> 4;

  v8f accf[4];
  #pragma unroll
  for (int f = 0; f < 4; ++f) accf[f] = (v8f){0, 0, 0, 0, 0, 0, 0, 0};

  #pragma unroll 1
  for (int c = 0; c < 26; ++c) {
    // ---- stage h chunk: 64 nodes x 64 k ----
    if (c < 10) {
      // unscaled regions: verbatim copies -> async path
      for (int i = tid; i < 64 * 16; i += 256) {
        int m  = i >> 4;
        int k4 = (i & 15) << 2;
        int node = min(n0 + m, N_NODES - 1);
        const float* p = (c < 2)
            ? &x[(size_t)node * IN_DIM + c * 64 + k4]
            : &agg[(size_t)node * 512 + (c - 2) * 64 + k4];
        cp16(&sA[m][k4], p);
      }
    } else {
      // scaled regions: load, scale, store
      for (int i = tid; i < 64 * 16; i += 256) {
        int m  = i >> 4;
        int k4 = (i & 15) << 2;
        int node = min(n0 + m, N_NODES - 1);
        float s;
        const float* p;
        if (c < 18) { p = &agg[(size_t)node * 512 + (c - 10) * 64 + k4]; s = sAmp[m]; }
        else        { p = &agg[(size_t)node * 512 + (c - 18) * 64 + k4]; s = sAtt[m]; }
        float4 v = *(const float4*)p;
        v.x *= s; v.y *= s; v.z *= s; v.w *= s;
        *(float4*)&sA[m][k4] = v;
      }
    }
    // ---- stage W1 chunk: interleaved pairs, verbatim (async 16B) ----
    for (int i = tid; i < 32 * 64; i += 256) {
      int kk = i >> 6;
      int q  = (i & 63) << 1;
      cp16(&sB2[kk][q], &W1p[(size_t)(c * 32 + kk) * 128 + q]);
    }
    if (c + 1 < 26)
      __builtin_prefetch(&W1p[(size_t)(c + 1) * 32 * 128 + tid * 16], 0, 1);
    cp_commit();
    __syncthreads();

    #pragma unroll
    for (int k0 = 0; k0 < 64; k0 += 4) {
      v2f a = *(const v2f*)&sA[mbase + ml][k0 + 2 * kp];
      int kkB = (k0 >> 1) + kp;
      #pragma unroll
      for (int f = 0; f < 4; ++f) {
        v2f b = *(const v2f*)&sB2[kkB][nbase + f * 16 + ml];
        accf[f] = wmma_f32x4(a, b, accf[f]);
      }
    }
    __syncthreads();
  }

  // ---- h1 = relu(acc + b1) -> raw view over sB2 storage (64x128, stride 132) ----
  float* sH1 = (float*)sB2;                        // 9216 floats >= 64*132
  #pragma unroll
  for (int f = 0; f < 4; ++f) {
    int n = nbase + f * 16 + ml;
    float bb = b1[n];
    #pragma unroll
    for (int r = 0; r < 8; ++r) {
      int m = mbase + r + 8 * kp;
      sH1[m * 132 + n] = fmaxf(accf[f][r] + bb, 0.0f);
    }
  }
  __syncthreads();

  // ---- GEMM2: h1(64x128) @ W2(128x128); B from interleaved W2 in L2 ----
  v8f acc2[4];
  #pragma unroll
  for (int f = 0; f < 4; ++f) acc2[f] = (v8f){0, 0, 0, 0, 0, 0, 0, 0};

  #pragma unroll 4
  for (int k0 = 0; k0 < 128; k0 += 4) {
    v2f a = *(const v2f*)&sH1[(mbase + ml) * 132 + k0 + 2 * kp];
    int kkB = (k0 >> 1) + kp;
    #pragma unroll
    for (int f = 0; f < 4; ++f) {
      int n = nbase + f * 16 + ml;
      v2f b = *(const v2f*)&W2p[(size_t)kkB * 128 + n];
      acc2[f] = wmma_f32x4(a, b, acc2[f]);
    }
  }

  // ---- epilogue: + b2 + residual ----
  #pragma unroll
  for (int f = 0; f < 4; ++f) {
    int n = nbase + f * 16 + ml;
    float bb = b2[n];
    #pragma unroll
    for (int r = 0; r < 8; ++r) {
      int m = mbase + r + 8 * kp;
      int node = n0 + m;
      if (node < N_NODES)
        out[(size_t)node * OUT_DIM + n] = acc2[f][r] + bb + x[(size_t)node * IN_DIM + n];
    }
  }
}

// ---------------------------------------------------------------------------
extern "C" void kernel_launch(void* const* d_in, const int* in_sizes, int n_in,
                              void* d_out, int out_size, void* d_ws, size_t ws_size,
                              hipStream_t stream) {
  const float* node_feat = (const float*)d_in[0];
  const float* edge_feat = (const float*)d_in[1];
  const int*   src       = (const int*)d_in[2];
  const int*   dst       = (const int*)d_in[3];
  const float* W_pre     = (const float*)d_in[4];
  const float* b_pre     = (const float*)d_in[5];
  const float* W_post1   = (const float*)d_in[6];
  const float* b_post1   = (const float*)d_in[7];
  const float* W_post2   = (const float*)d_in[8];
  const float* b_post2   = (const float*)d_in[9];
  float* out = (float*)d_out;

  // workspace: acc [N][4][128] f32, deg/amp/att [N] each, then interleaved
  // weight copies (float2 K-pairs). ~104 MB total.
  float* acc = (float*)d_ws;
  float* deg = acc + (size_t)N_NODES * 512;
  float* amp = deg + N_NODES;
  float* att = amp + N_NODES;
  float2* Wpre2 = (float2*)(att + N_NODES);       // 160*128 float2
  float2* W1p   = Wpre2 + (size_t)160 * 128;      // 832*128 float2
  float2* W2p   = W1p + (size_t)832 * 128;        // 64*128 float2

  pna_interleave_w<<<(160 * 128 + 255) / 256, 256, 0, stream>>>(W_pre, Wpre2, 320);
  pna_interleave_w<<<(832 * 128 + 255) / 256, 256, 0, stream>>>(W_post1, W1p, 1664);
  pna_interleave_w<<<(64 * 128 + 255) / 256, 256, 0, stream>>>(W_post2, W2p, 128);

  {
    long total = (long)N_NODES * 512;
    int blocks = (int)((total + 255) / 256);
    pna_init<<<blocks, 256, 0, stream>>>((unsigned*)acc, deg);
  }
  pna_edge<<<N_EDGES / 64, 256, 0, stream>>>(node_feat, edge_feat, src, dst,
                                             Wpre2, b_pre, acc, deg);
  pna_finalize<<<(N_NODES * 128) / 256, 256, 0, stream>>>(acc, deg, amp, att);
  pna_node<<<(N_NODES + 63) / 64, 256, 0, stream>>>(node_feat, acc, amp, att,
                                                    W1p, b_post1, W2p, b_post2,
                                                    out);
}